// PatchSlicedTransport_37151467110990
// MI455X (gfx1250) — compile-verified
//
#include <hip/hip_runtime.h>

// ---------------- problem constants (match reference) ----------------
#define NDATA  4096
#define DDIM   12288          // 64*64*3
#define NPATCH 768
#define PDIM   16
#define NCOMP  16
#define NBIN   200
#define TDIM   (NPATCH * NCOMP)

#define WAVES_PER_BLOCK   8
#define PATCHES_PER_WAVE  12
#define PATCH_GROUPS      (NPATCH / (WAVES_PER_BLOCK * PATCHES_PER_WAVE))   // 8
#define LROW 20   // LDS row stride in words: halves land in disjoint bank ranges mod 64

typedef __attribute__((ext_vector_type(2))) float v2f;
typedef __attribute__((ext_vector_type(8))) float v8f;

__global__ void __launch_bounds__(256)
zero_f32_kernel(float* __restrict__ p, int n) {
    int i = blockIdx.x * blockDim.x + threadIdx.x;
    if (i < n) p[i] = 0.0f;
}

__global__ void __launch_bounds__(256)
patch_spline_transport_kernel(const float* __restrict__ data,
                              const float* __restrict__ wT,
                              const float* __restrict__ kx,
                              const float* __restrict__ ky,
                              const float* __restrict__ kd,
                              float* __restrict__ out,
                              float* __restrict__ logdet)
{
    __shared__ float bufA[WAVES_PER_BLOCK][16 * LROW];   // staged input tile (m,k)
    __shared__ float bufD[WAVES_PER_BLOCK][16 * LROW];   // delta tile (m,c) for transpose

    const int lane = threadIdx.x & 31;
    const int wave = threadIdx.x >> 5;
    const int half = lane >> 4;      // 0: lanes 0-15, 1: lanes 16-31
    const int l16  = lane & 15;

    const int n0 = blockIdx.x * 16;  // 16-sample tile base
    float* A_lds = bufA[wave];
    float* D_lds = bufD[wave];

    float acc[8];
    #pragma unroll
    for (int r = 0; r < 8; ++r) acc[r] = 0.0f;

    const int pbase = blockIdx.y * (WAVES_PER_BLOCK * PATCHES_PER_WAVE)
                    + wave * PATCHES_PER_WAVE;

    for (int pi = 0; pi < PATCHES_PER_WAVE; ++pi) {
        const int p   = pbase + pi;
        const int hh  = p / 48;                 // patch index = h*48 + w*3 + c
        const int rem = p - hh * 48;
        const int ww  = rem / 3;
        const int cc  = rem - ww * 3;
        const int dbase = hh * 768 + ww * 12 + cc;   // flat offset of (kh=0,kw=0)

        // ---------- load A tile: lane holds sample m=l16; K = 4j + v + 2*half ----------
        const float* drow = data + (size_t)(n0 + l16) * DDIM + dbase;
        v2f a[4];
        #pragma unroll
        for (int j = 0; j < 4; ++j) {
            #pragma unroll
            for (int v = 0; v < 2; ++v) {
                const int k  = 4 * j + v + 2 * half;
                const int kh = k >> 2, kw = k & 3;
                const float x = drow[kh * 192 + kw * 3];
                a[j][v] = x;
                A_lds[l16 * LROW + k] = x;       // park for the final data+delta store
            }
        }

        // software prefetch of next patch's input row (L2 warm-up)
        if (pi + 1 < PATCHES_PER_WAVE) {
            const int pn = p + 1;
            const int hn = pn / 48, remn = pn - hn * 48;
            const int wn = remn / 3, cn = remn - wn * 3;
            __builtin_prefetch(data + (size_t)(n0 + l16) * DDIM + hn * 768 + wn * 12 + cn, 0, 3);
        }

        // ---------- B = wT[p] : (K=k) x (N=c) ----------
        const float* wp = wT + p * (PDIM * NCOMP);
        v2f b[4];
        #pragma unroll
        for (int j = 0; j < 4; ++j)
            #pragma unroll
            for (int v = 0; v < 2; ++v) {
                const int k = 4 * j + v + 2 * half;
                b[j][v] = wp[k * NCOMP + l16];
            }

        // ---------- proj tile = A(16x16) * B(16x16) via 4 chained f32 WMMAs ----------
        v8f c = {};
        #pragma unroll
        for (int j = 0; j < 4; ++j)
            c = __builtin_amdgcn_wmma_f32_16x16x4_f32(false, a[j], false, b[j],
                                                      (short)0, c, false, false);
        // c: VGPR r, lane -> sample m = r + 8*half, component = l16

        // ---------- RQ spline: one knot row per lane (t = p*16 + comp) ----------
        const int t = p * NCOMP + l16;
        const float* xrow = kx + (size_t)t * NBIN;
        const float* yrow = ky + (size_t)t * NBIN;
        const float* wrow = kd + (size_t)t * NBIN;

        #pragma unroll
        for (int r = 0; r < 8; ++r) {
            const float X = c[r];
            // searchsorted 'left': count of knots < X (branchless binary search)
            int pos = 0;
            #pragma unroll
            for (int s = 128; s; s >>= 1) {
                const int np = pos + s;
                if (np <= NBIN && xrow[np - 1] < X) pos = np;
            }
            int kb = pos - 1;
            kb = kb < 0 ? 0 : (kb > NBIN - 2 ? NBIN - 2 : kb);

            const float xk  = xrow[kb], xk1 = xrow[kb + 1];
            const float yk  = yrow[kb], yk1 = yrow[kb + 1];
            const float dk  = wrow[kb], dk1 = wrow[kb + 1];

            const float rwx = __builtin_amdgcn_rcpf(xk1 - xk);
            const float dyk = yk1 - yk;
            const float sl  = dyk * rwx;
            float xi = (X - xk) * rwx;
            xi = xi < 0.0f ? 0.0f : (xi > 1.0f ? 1.0f : xi);
            const float xi1  = 1.0f - xi;
            const float xixi1 = xi * xi1;
            const float den  = sl + (dk + dk1 - 2.0f * sl) * xixi1;
            const float rden = __builtin_amdgcn_rcpf(den);
            const float y_in = yk + dyk * (sl * xi * xi + dk * xixi1) * rden;
            const float sd   = sl * rden;
            const float dy_in = sd * sd * (dk1 * xi * xi + 2.0f * sl * xixi1 + dk * xi1 * xi1);

            const bool below = X < xk;     // only possible when kb==0
            const bool above = X > xk1;    // only possible when kb==NBIN-2
            const float y  = below ? (yk  + (X - xk ) * dk )
                           : above ? (yk1 + (X - xk1) * dk1) : y_in;
            const float dy = below ? dk : (above ? dk1 : dy_in);

            acc[r] += __logf(dy);
            const int m = r + 8 * half;
            D_lds[m * LROW + l16] = y - X;          // delta in (sample, comp) layout
        }

        // ---------- second GEMM: dpatch = delta(16x16, K=comp) * wT[p]^T ----------
        v2f a2[4], b2[4];
        #pragma unroll
        for (int j = 0; j < 4; ++j)
            #pragma unroll
            for (int v = 0; v < 2; ++v) {
                const int kq = 4 * j + v + 2 * half;
                a2[j][v] = D_lds[l16 * LROW + kq];    // transpose via LDS
                b2[j][v] = wp[l16 * NCOMP + kq];      // (K=comp=kq, N=k=l16)
            }
        v8f c2 = {};
        #pragma unroll
        for (int j = 0; j < 4; ++j)
            c2 = __builtin_amdgcn_wmma_f32_16x16x4_f32(false, a2[j], false, b2[j],
                                                       (short)0, c2, false, false);

        // ---------- scatter: out = data + dpatch (each (n,d) owned by one patch) ----------
        const int kh = l16 >> 2, kw = l16 & 3;
        const int dcol = dbase + kh * 192 + kw * 3;
        #pragma unroll
        for (int r = 0; r < 8; ++r) {
            const int m = r + 8 * half;
            out[(size_t)(n0 + m) * DDIM + dcol] = A_lds[m * LROW + l16] + c2[r];
        }
    }

    // ---------- per-sample logdet: butterfly over each 16-lane half, one atomic ----------
    #pragma unroll
    for (int r = 0; r < 8; ++r) {
        float v = acc[r];
        v += __shfl_xor(v, 1, 32);
        v += __shfl_xor(v, 2, 32);
        v += __shfl_xor(v, 4, 32);
        v += __shfl_xor(v, 8, 32);
        if (l16 == 0) atomicAdd(&logdet[n0 + r + 8 * half], v);
    }
}

extern "C" void kernel_launch(void* const* d_in, const int* in_sizes, int n_in,
                              void* d_out, int out_size, void* d_ws, size_t ws_size,
                              hipStream_t stream) {
    const float* data = (const float*)d_in[0];
    const float* wT   = (const float*)d_in[1];
    const float* kxp  = (const float*)d_in[2];
    const float* kyp  = (const float*)d_in[3];
    const float* kdp  = (const float*)d_in[4];

    float* out    = (float*)d_out;
    float* logdet = out + (size_t)NDATA * DDIM;   // outputs concatenated flat

    hipLaunchKernelGGL(zero_f32_kernel, dim3((NDATA + 255) / 256), dim3(256), 0, stream,
                       logdet, NDATA);

    dim3 grid(NDATA / 16, PATCH_GROUPS);          // 256 x 8 blocks, 8 waves each
    hipLaunchKernelGGL(patch_spline_transport_kernel, grid, dim3(256), 0, stream,
                       data, wT, kxp, kyp, kdp, out, logdet);
}